// HyperConnections_4166118277386
// MI455X (gfx1250) — compile-verified
//
#include <hip/hip_runtime.h>
#include <hip/hip_bf16.h>
#include <math.h>

// ---------------------------------------------------------------------------
// HyperConnections fused kernel for MI455X (gfx1250, wave32).
//
// Per token the reference collapses to:
//   invr      = rsqrt(mean(x^2) + 1e-5)
//   P[i]      = sum_d (x*w)[d] * dynamic_alpha_fn[d,i]      (i = 0..E)
//   Pbeta     = sum_d (x*w)[d] * dynamic_beta_fn[d]
//   beta_sum  = sum(static_beta) + E * tanh(invr*Pbeta) * beta_scale
//   coeff_sum = sum_{i=1..E} colsum(static_alpha)[i]
//             + E * sum_{i=1..E} tanh(invr*P[i]) * alpha_scale
//   out       = residual*beta_sum + x*coeff_sum
//
// Bandwidth bound: ~402 MB minimum traffic -> ~17 us at 23.3 TB/s.
// The [16 tokens x 2048] @ [2048 x 6] projection is done with
// V_WMMA_F32_16X16X4_F32 (fp32 precision), 8 waves splitting K.
// ---------------------------------------------------------------------------

typedef float v8f __attribute__((ext_vector_type(8)));
typedef float v2f __attribute__((ext_vector_type(2)));

#define DM      2048         // D_MODEL
#define EE      4            // EXP_RATE
#define TPB     16           // tokens per block (WMMA M dimension)
#define CHUNK   256          // K chunk staged in LDS
#define THREADS 256          // 8 waves (wave32)
#define WAVES   8
#define TROW    (CHUNK + 4)  // row pad: stride%64==4 -> conflict-free b64 A reads

__global__ __launch_bounds__(THREADS)
void hyperconn_fused_kernel(const float* __restrict__ x,
                            const float* __restrict__ residual,
                            const float* __restrict__ w,
                            const float* __restrict__ sAlpha,   // [E, E+1]
                            const float* __restrict__ sBeta,    // [E]
                            const float* __restrict__ dAfn,     // [D, E+1]
                            const float* __restrict__ aScale,   // [1]
                            const float* __restrict__ dBfn,     // [D]
                            const float* __restrict__ bScale,   // [1]
                            float* __restrict__ out)
{
    __shared__ float t_lds[TPB * TROW];          // (x*w) tile, 16x256 padded
    __shared__ float bl[(CHUNK / 4) * 16 * 4];   // B tile: bl[kq][n][j] = Bm[kq*4+j][n]
    __shared__ float wacc[WAVES * 8 * 32];       // per-wave WMMA partial D tiles
    __shared__ float ssq[TPB * 16];              // sum(x^2) partials
    __shared__ float pmat[TPB][8];               // P[token][col 0..5]
    __shared__ float bsArr[TPB], csArr[TPB];     // final per-token scalars

    const int tid  = threadIdx.x;
    const int wave = tid >> 5;
    const int lane = tid & 31;
    const int m16  = lane & 15;      // WMMA row / column index within lane half
    const int hh   = lane >> 4;      // lane half selects K {0,1} vs {2,3}

    const long tokBase = (long)blockIdx.x * TPB;
    const float* xTile = x + tokBase * DM;

    // Loader mapping: thread (lrow, lslot) covers row lrow, 4 float4s per chunk.
    const int lrow  = tid >> 4;      // 0..15 token row
    const int lslot = tid & 15;

    v8f acc = {};                    // 16x16 f32 D accumulator (tokens x cols)
    float ssqAcc = 0.f;

    for (int kb = 0; kb < DM; kb += CHUNK) {
        // ---- stage t = x*w (coalesced: lanes read consecutive float4s) ----
        #pragma unroll
        for (int j = 0; j < 4; ++j) {
            const int kl = (j * 16 + lslot) * 4;           // 0..255
            const float4 xv = *(const float4*)(xTile + (long)lrow * DM + kb + kl);
            const float4 wv = *(const float4*)(w + kb + kl);
            ssqAcc += xv.x * xv.x + xv.y * xv.y + xv.z * xv.z + xv.w * xv.w;
            float* t = &t_lds[lrow * TROW + kl];
            t[0] = xv.x * wv.x; t[1] = xv.y * wv.y;
            t[2] = xv.z * wv.z; t[3] = xv.w * wv.w;
        }
        // ---- stage padded B tile: cols 0..4 = dAfn, col 5 = dBfn, rest 0 ----
        for (int e = tid; e < (CHUNK / 4) * 64; e += THREADS) {
            const int kq = e >> 6, rem = e & 63, n = rem >> 2, j = rem & 3;
            const int kg = kb + kq * 4 + j;
            float v = 0.f;
            if (n < EE + 1)       v = dAfn[(long)kg * (EE + 1) + n];
            else if (n == EE + 1) v = dBfn[kg];
            bl[e] = v;
        }
        __syncthreads();

        // ---- WMMA: wave 'wave' covers K sub-range [wave*32, wave*32+32) ----
        // A 16x4 f32 layout: lane L -> row L%16; VGPR0/1 = K {0,1} (L<16) or {2,3}.
        // B  4x16 f32 layout mirrored: lane L -> col L%16; VGPR0/1 same K split.
        #pragma unroll
        for (int i = 0; i < 8; ++i) {
            const int kq = wave * 8 + i;
            const int k0 = kq * 4;
            v2f a = *(const v2f*)&t_lds[m16 * TROW + k0 + 2 * hh];
            v2f b = *(const v2f*)&bl[kq * 64 + m16 * 4 + 2 * hh];
            acc = __builtin_amdgcn_wmma_f32_16x16x4_f32(
                false, a, false, b, (short)0, acc, false, false);
        }
        __syncthreads();
    }

    // ---- reduce per-wave partial D tiles + sumsq partials ----
    #pragma unroll
    for (int r = 0; r < 8; ++r) wacc[(wave * 8 + r) * 32 + lane] = acc[r];
    ssq[lrow * 16 + lslot] = ssqAcc;
    __syncthreads();

    {
        const int r = tid >> 5, ln = tid & 31;
        float s = 0.f;
        #pragma unroll
        for (int wv = 0; wv < WAVES; ++wv) s += wacc[(wv * 8 + r) * 32 + ln];
        // D layout: VGPR r -> M=r (lanes 0-15) or M=r+8 (lanes 16-31), N=lane%16
        const int mrow = r + ((ln >> 4) << 3);
        const int n    = ln & 15;
        if (n < EE + 2) pmat[mrow][n] = s;
    }
    __syncthreads();

    if (tid < TPB) {
        float sq = 0.f;
        #pragma unroll
        for (int s = 0; s < 16; ++s) sq += ssq[tid * 16 + s];
        const float invr = rsqrtf(sq * (1.0f / DM) + 1e-5f);
        const float as = aScale[0], bsc = bScale[0];

        const float dynB = tanhf(invr * pmat[tid][EE + 1]) * bsc;
        float betaSum = (float)EE * dynB;
        #pragma unroll
        for (int e = 0; e < EE; ++e) betaSum += sBeta[e];

        float coeff = 0.f;
        #pragma unroll
        for (int i = 1; i <= EE; ++i) {
            float col = 0.f;
            #pragma unroll
            for (int jj = 0; jj < EE; ++jj) col += sAlpha[jj * (EE + 1) + i];
            coeff += col + (float)EE * (tanhf(invr * pmat[tid][i]) * as);
        }
        bsArr[tid] = betaSum;
        csArr[tid] = coeff;
    }
    __syncthreads();

    // ---- streaming AXPY: out = residual*beta_sum + x*coeff_sum ----
    const float4* xv4 = (const float4*)xTile;                 // L2-resident re-read
    const float4* rv4 = (const float4*)(residual + tokBase * DM);
    float4*       ov4 = (float4*)(out + tokBase * DM);
    for (int i4 = tid; i4 < TPB * (DM / 4); i4 += THREADS) {
        const int   row = i4 >> 9;                            // DM/4 == 512
        const float bsv = bsArr[row], csv = csArr[row];
        const float4 xv = xv4[i4];
        const float4 rv = rv4[i4];
        float4 o;
        o.x = rv.x * bsv + xv.x * csv;
        o.y = rv.y * bsv + xv.y * csv;
        o.z = rv.z * bsv + xv.z * csv;
        o.w = rv.w * bsv + xv.w * csv;
        ov4[i4] = o;
    }
}

extern "C" void kernel_launch(void* const* d_in, const int* in_sizes, int n_in,
                              void* d_out, int out_size, void* d_ws, size_t ws_size,
                              hipStream_t stream) {
    const float* x      = (const float*)d_in[0];
    const float* resid  = (const float*)d_in[1];
    const float* rmsw   = (const float*)d_in[2];
    const float* sAlpha = (const float*)d_in[3];
    const float* sBeta  = (const float*)d_in[4];
    const float* dAfn   = (const float*)d_in[5];
    const float* aScale = (const float*)d_in[6];
    const float* dBfn   = (const float*)d_in[7];
    const float* bScale = (const float*)d_in[8];
    float* out = (float*)d_out;

    const int numTokens = in_sizes[0] / DM;      // B*T = 16384
    const int blocks = numTokens / TPB;          // 1024 workgroups of 16 tokens

    hyperconn_fused_kernel<<<blocks, THREADS, 0, stream>>>(
        x, resid, rmsw, sAlpha, sBeta, dAfn, aScale, dBfn, bScale, out);
}